// DGCNN_exp_39900246180144
// MI455X (gfx1250) — compile-verified
//
#include <hip/hip_runtime.h>
#include <stdint.h>

#define BEPS 1e-5f
#define Bn   4
#define Cin  1000
#define Nn   4096
#define NK   (3*Nn)

typedef __attribute__((ext_vector_type(2))) float v2f;
typedef __attribute__((ext_vector_type(8))) float v8f;

static __device__ __forceinline__ v8f wmma4(v2f a, v2f b, v8f c) {
    // D = A(16x4,f32) * B(4x16,f32) + C(16x16,f32)
    return __builtin_amdgcn_wmma_f32_16x16x4_f32(false, a, false, b, (short)0, c, false, false);
}

// ---------------------------------------------------------------- norms
__global__ void norms_kernel(const float* __restrict__ x, float* __restrict__ xx) {
    int gid = blockIdx.x * 256 + threadIdx.x;          // B*N threads
    int b = gid >> 12, n = gid & (Nn - 1);
    const float* xb = x + (size_t)b * Cin * Nn + n;
    float s = 0.f;
    for (int c = 0; c < Cin; ++c) { float v = xb[(size_t)c * Nn]; s += v * v; }
    xx[gid] = s;
}

// ------------------------------------------------- KNN: WMMA GEMM + fused top-3
// One wave = 16 rows x all 4096 cols (in 128-col blocks). blockDim=128 (4 waves).
__global__ void __launch_bounds__(128) knn_kernel(const float* __restrict__ x,
                                                  const float* __restrict__ xx,
                                                  int* __restrict__ idx) {
    __shared__ float stage[4 * 16 * 128];
    __shared__ float mv[4][32][3];
    __shared__ int   mi[4][32][3];

    const int tid  = threadIdx.x;
    const int wave = tid >> 5, lane = tid & 31;
    const int m    = lane & 15, half = lane >> 4;
    const int gw   = blockIdx.x * 4 + wave;            // 1024 waves total
    const int b    = gw >> 8;                          // 256 row-blocks / batch
    const int iBase = (gw & 255) * 16;

    const float* xb  = x  + (size_t)b * Cin * Nn;
    const float* xxb = xx + (size_t)b * Nn;
    float* st = stage + wave * 16 * 128;

    const v8f zero8 = {0.f,0.f,0.f,0.f,0.f,0.f,0.f,0.f};
    float tv0 = -__builtin_inff(), tv1 = tv0, tv2 = tv0;
    int   ti0 = 0, ti1 = 0, ti2 = 0;
    const float xxi = xxb[iBase + m];

    for (int jBase = 0; jBase < Nn; jBase += 128) {
        v8f acc[8];
        #pragma unroll
        for (int t = 0; t < 8; ++t) acc[t] = zero8;

        for (int c = 0; c < Cin; c += 4) {
            const float* pk = xb + (size_t)(c + 2 * half) * Nn;
            v2f av; av.x = pk[iBase + m]; av.y = pk[Nn + iBase + m];
            const float* pb = pk + jBase + m;
            #pragma unroll
            for (int t = 0; t < 8; ++t) {
                v2f bv; bv.x = pb[t * 16]; bv.y = pb[Nn + t * 16];
                acc[t] = wmma4(av, bv, acc[t]);
            }
        }
        __builtin_amdgcn_wave_barrier();
        #pragma unroll
        for (int t = 0; t < 8; ++t)
            #pragma unroll
            for (int r = 0; r < 8; ++r)
                st[(r + 8 * half) * 128 + t * 16 + m] = acc[t][r];
        __builtin_amdgcn_wave_barrier();

        // lane scans row m, 64-col segment selected by `half`
        const float* row = st + m * 128 + half * 64;
        const int j0col = jBase + half * 64;
        for (int cc = 0; cc < 64; ++cc) {
            float pd = 2.f * row[cc] - xxi - xxb[j0col + cc];
            int j = j0col + cc;
            if (pd > tv2) {
                if (pd > tv1) {
                    if (pd > tv0) { tv2=tv1; ti2=ti1; tv1=tv0; ti1=ti0; tv0=pd; ti0=j; }
                    else          { tv2=tv1; ti2=ti1; tv1=pd;  ti1=j; }
                } else            { tv2=pd;  ti2=j; }
            }
        }
        __builtin_amdgcn_wave_barrier();
    }

    mv[wave][lane][0] = tv0; mv[wave][lane][1] = tv1; mv[wave][lane][2] = tv2;
    mi[wave][lane][0] = ti0; mi[wave][lane][1] = ti1; mi[wave][lane][2] = ti2;
    __builtin_amdgcn_wave_barrier();

    if (half == 0) {
        float cv[6] = { tv0, tv1, tv2,
                        mv[wave][lane+16][0], mv[wave][lane+16][1], mv[wave][lane+16][2] };
        int   ci[6] = { ti0, ti1, ti2,
                        mi[wave][lane+16][0], mi[wave][lane+16][1], mi[wave][lane+16][2] };
        int* op = idx + ((size_t)b * Nn + iBase + m) * 3;
        #pragma unroll
        for (int s = 0; s < 3; ++s) {
            int best = 0; float bvv = cv[0]; int bii = ci[0];
            #pragma unroll
            for (int t = 1; t < 6; ++t)
                if (cv[t] > bvv || (cv[t] == bvv && ci[t] < bii)) { bvv = cv[t]; bii = ci[t]; best = t; }
            op[s] = bii;
            cv[best] = -__builtin_inff();
        }
    }
}

// --------------------------------------------- generic WMMA GEMM (+BN/ReLU/max)
// Y(Co x NC) = W(Co x Ci) * H(Ci x NC) per batch. One wave: 16 rows x 64 cols.
template <bool EPI, bool WRITE, bool DOMAX>
__global__ void __launch_bounds__(256) gemm_kernel(
        const float* __restrict__ W, const float* __restrict__ H,
        float* __restrict__ Y, float* __restrict__ XM,
        const float* __restrict__ g,  const float* __restrict__ bb,
        const float* __restrict__ mm, const float* __restrict__ vv,
        int Ci, int NC, int rowTiles, int nTiles64,
        long long strideH, long long strideY, long long strideX) {
    const int tid = threadIdx.x;
    const int lane = tid & 31, wave = tid >> 5;
    const int m = lane & 15, half = lane >> 4;
    const int gw = blockIdx.x * 8 + wave;
    const int wpb = rowTiles * nTiles64;
    const int b = gw / wpb;
    int rem = gw - b * wpb;
    const int rowTile = rem / nTiles64;
    const int colTile = rem - rowTile * nTiles64;
    const int oBase = rowTile * 16, nBase = colTile * 64;

    const float* Hb = H + (size_t)b * strideH;
    const v2f*   pA = (const v2f*)(W + (size_t)(oBase + m) * Ci + 2 * half);

    const v8f zero8 = {0.f,0.f,0.f,0.f,0.f,0.f,0.f,0.f};
    v8f a0 = zero8, a1 = zero8, a2 = zero8, a3 = zero8;

    for (int c = 0; c < Ci; c += 4) {
        v2f a = pA[c >> 1];
        const float* p0 = Hb + (size_t)(c + 2 * half) * NC + nBase + m;
        v2f b0; b0.x = p0[0];  b0.y = p0[NC];
        v2f b1; b1.x = p0[16]; b1.y = p0[NC + 16];
        v2f b2; b2.x = p0[32]; b2.y = p0[NC + 32];
        v2f b3; b3.x = p0[48]; b3.y = p0[NC + 48];
        a0 = wmma4(a, b0, a0); a1 = wmma4(a, b1, a1);
        a2 = wmma4(a, b2, a2); a3 = wmma4(a, b3, a3);
    }

    float* Yb = WRITE ? (Y  + (size_t)b * strideY) : nullptr;
    float* Xb = DOMAX ? (XM + (size_t)b * strideX) : nullptr;

    #pragma unroll
    for (int r = 0; r < 8; ++r) {
        const int o = oBase + r + 8 * half;
        float scale = 1.f, bias = 0.f;
        if constexpr (EPI) {
            float sc = g[o] * rsqrtf(vv[o] + BEPS);
            scale = sc; bias = bb[o] - mm[o] * sc;
        }
        float vals[4] = { a0[r], a1[r], a2[r], a3[r] };
        #pragma unroll
        for (int t = 0; t < 4; ++t) {
            const int col = nBase + t * 16 + m;
            float y = vals[t];
            if constexpr (EPI) y = fmaxf(y * scale + bias, 0.f);
            if constexpr (WRITE) Yb[(size_t)o * NC + col] = y;
            if constexpr (DOMAX) {
                const int n = col & (Nn - 1);   // col = k*N + n
                atomicMax((unsigned int*)(Xb + (size_t)o * Nn + n), __float_as_uint(y));
            }
        }
    }
}

// --------------------------------------------------------------- w1 repack
// W'(128x1000): rows 0..63 = w1[:, :1000] (neighbor), rows 64..127 = w1[:, 1000:]
__global__ void repack_w1_kernel(const float* __restrict__ w1, float* __restrict__ wp) {
    int gid = blockIdx.x * 256 + threadIdx.x;          // 128000 threads
    int r = gid / 1000, c = gid - r * 1000;
    wp[gid] = (r < 64) ? w1[r * 2000 + c] : w1[(r - 64) * 2000 + 1000 + c];
}

// ------------------------------------ conv1 epilogue: gather + BN/ReLU + max_k
__global__ void feat1_kernel(const float* __restrict__ pq, const int* __restrict__ idx,
                             const float* __restrict__ g,  const float* __restrict__ bb,
                             const float* __restrict__ mm, const float* __restrict__ vv,
                             float* __restrict__ h1, float* __restrict__ xcat) {
    int gid = blockIdx.x * 256 + threadIdx.x;          // B*N threads
    int b = gid >> 12, n = gid & (Nn - 1);
    const float* P = pq + (size_t)b * 128 * Nn;        // rows 0..63  : w1A @ x
    const float* Q = P + (size_t)64 * Nn;              // rows 64..127: w1B @ x
    const int* ip = idx + (size_t)gid * 3;
    int j0 = ip[0], j1 = ip[1], j2 = ip[2];
    float* h1b = h1   + (size_t)b * 64 * 3 * Nn + n;   // (B,64,3,N)
    float* xc  = xcat + (size_t)b * 960 * Nn + n;      // rows 0..63 of xcat
    for (int o = 0; o < 64; ++o) {
        float sc = g[o] * rsqrtf(vv[o] + BEPS);
        float bi = bb[o] - mm[o] * sc;
        const float* Po = P + (size_t)o * Nn;
        float q = Q[(size_t)o * Nn + n];
        float y0 = fmaxf((Po[j0] + q) * sc + bi, 0.f);
        float y1 = fmaxf((Po[j1] + q) * sc + bi, 0.f);
        float y2 = fmaxf((Po[j2] + q) * sc + bi, 0.f);
        h1b[(size_t)o * 3 * Nn]           = y0;
        h1b[(size_t)o * 3 * Nn + Nn]      = y1;
        h1b[(size_t)o * 3 * Nn + 2 * Nn]  = y2;
        xc[(size_t)o * Nn] = fmaxf(y0, fmaxf(y1, y2));
    }
}

// ----------------------------------------------------------------- launch
extern "C" void kernel_launch(void* const* d_in, const int* in_sizes, int n_in,
                              void* d_out, int out_size, void* d_ws, size_t ws_size,
                              hipStream_t stream) {
    const float* x = (const float*)d_in[0];
    const float *w[5], *g[5], *bp[5], *mp[5], *vp[5];
    for (int i = 0; i < 5; ++i) {
        w[i]  = (const float*)d_in[1 + 5 * i];
        g[i]  = (const float*)d_in[2 + 5 * i];
        bp[i] = (const float*)d_in[3 + 5 * i];
        mp[i] = (const float*)d_in[4 + 5 * i];
        vp[i] = (const float*)d_in[5 + 5 * i];
    }
    float* ws = (float*)d_ws;
    size_t off = 0;
    float* xx   = ws + off;        off += (size_t)Bn * Nn;
    int*   idx  = (int*)(ws + off); off += (size_t)Bn * Nn * 3;
    float* wp   = ws + off;        off += (size_t)128 * 1000;
    float* pq   = ws + off;        off += (size_t)Bn * 128 * Nn;
    float* h1   = ws + off;        off += (size_t)Bn * 64 * NK;
    float* h2   = ws + off;        off += (size_t)Bn * 128 * NK;
    float* xcat = ws + off;        off += (size_t)Bn * 960 * Nn;
    float* h3   = (float*)d_out;   // scratch inside d_out; conv5 overwrites it
    float* out  = (float*)d_out;

    hipMemsetAsync(xcat, 0, (size_t)Bn * 960 * Nn * sizeof(float), stream);
    repack_w1_kernel<<<500, 256, 0, stream>>>(w[0], wp);
    norms_kernel<<<Bn * Nn / 256, 256, 0, stream>>>(x, xx);
    knn_kernel<<<Bn * (Nn / 16) / 4, 128, 0, stream>>>(x, xx, idx);

    // PQ = [w1A;w1B](128x1000) @ x(1000x4096) — raw output
    gemm_kernel<false, true, false><<<Bn * 8 * 64 / 8, 256, 0, stream>>>(
        wp, x, pq, nullptr, nullptr, nullptr, nullptr, nullptr,
        Cin, Nn, 8, 64, (long long)Cin * Nn, (long long)128 * Nn, 0LL);

    feat1_kernel<<<Bn * Nn / 256, 256, 0, stream>>>(pq, idx, g[0], bp[0], mp[0], vp[0], h1, xcat);

    // conv2: 128x64 @ 64x12288 -> h2, max -> xcat rows 64..191
    gemm_kernel<true, true, true><<<Bn * 8 * 192 / 8, 256, 0, stream>>>(
        w[1], h1, h2, xcat + (size_t)64 * Nn, g[1], bp[1], mp[1], vp[1],
        64, NK, 8, 192, (long long)64 * NK, (long long)128 * NK, (long long)960 * Nn);

    // conv3: 256x128 @ 128x12288 -> h3 (in d_out), max -> xcat rows 192..447
    gemm_kernel<true, true, true><<<Bn * 16 * 192 / 8, 256, 0, stream>>>(
        w[2], h2, h3, xcat + (size_t)192 * Nn, g[2], bp[2], mp[2], vp[2],
        128, NK, 16, 192, (long long)128 * NK, (long long)256 * NK, (long long)960 * Nn);

    // conv4: 512x256 @ 256x12288 -> max only -> xcat rows 448..959 (h never stored)
    gemm_kernel<true, false, true><<<Bn * 32 * 192 / 8, 256, 0, stream>>>(
        w[3], h3, nullptr, xcat + (size_t)448 * Nn, g[3], bp[3], mp[3], vp[3],
        256, NK, 32, 192, (long long)256 * NK, 0LL, (long long)960 * Nn);

    // conv5: 1024x960 @ 960x4096 -> d_out
    gemm_kernel<true, true, false><<<Bn * 64 * 64 / 8, 256, 0, stream>>>(
        w[4], xcat, out, nullptr, g[4], bp[4], mp[4], vp[4],
        960, Nn, 64, 64, (long long)960 * Nn, (long long)1024 * Nn, 0LL);
}